// GraphSAGE_node_45801531245071
// MI455X (gfx1250) — compile-verified
//
#include <hip/hip_runtime.h>
#include <stdint.h>

#define N_NODES 50000
#define N_EDGES 1600000
#define DIM 128
#define BN_EPS 1e-5f

typedef __attribute__((ext_vector_type(2))) float v2f;
typedef __attribute__((ext_vector_type(4))) float v4f;
typedef __attribute__((ext_vector_type(8))) float v8f;

// ---------------- CSR construction ----------------

__global__ void zero_i32_kernel(int* __restrict__ p, int n) {
    int i = blockIdx.x * blockDim.x + threadIdx.x;
    if (i < n) p[i] = 0;
}

__global__ void zero_f32_kernel(float* __restrict__ p, int n) {
    int i = blockIdx.x * blockDim.x + threadIdx.x;
    if (i < n) p[i] = 0.0f;
}

__global__ void count_deg_kernel(const long long* __restrict__ dst, int* __restrict__ deg) {
    int e = blockIdx.x * blockDim.x + threadIdx.x;
    if (e < N_EDGES) atomicAdd(&deg[(int)dst[e]], 1);
}

// Single-block chunked inclusive scan -> exclusive row_ptr (N+1 entries).
__global__ void exscan_kernel(const int* __restrict__ deg, int* __restrict__ rowptr, int n) {
    __shared__ int buf[1024];
    __shared__ int s_carry;
    int tid = threadIdx.x;
    if (tid == 0) s_carry = 0;
    __syncthreads();
    for (int base = 0; base < n; base += 1024) {
        int v = (base + tid < n) ? deg[base + tid] : 0;
        int val = v;
        buf[tid] = val;
        __syncthreads();
        for (int off = 1; off < 1024; off <<= 1) {
            int t = (tid >= off) ? buf[tid - off] : 0;
            __syncthreads();
            val += t;
            buf[tid] = val;
            __syncthreads();
        }
        int carry = s_carry;                       // safe: sync'd since last write
        if (base + tid < n) rowptr[base + tid] = carry + val - v;  // exclusive
        __syncthreads();
        if (tid == 1023) s_carry = carry + val;    // chunk total
        __syncthreads();
    }
    if (tid == 0) rowptr[n] = s_carry;
}

__global__ void init_cursor_kernel(const int* __restrict__ rowptr, int* __restrict__ cursor) {
    int i = blockIdx.x * blockDim.x + threadIdx.x;
    if (i < N_NODES) cursor[i] = rowptr[i];
}

__global__ void scatter_edges_kernel(const long long* __restrict__ src,
                                     const long long* __restrict__ dst,
                                     int* __restrict__ cursor, int* __restrict__ cols) {
    int e = blockIdx.x * blockDim.x + threadIdx.x;
    if (e < N_EDGES) {
        int d = (int)dst[e];
        int p = atomicAdd(&cursor[d], 1);
        cols[p] = (int)src[e];
    }
}

// ---------------- Mean aggregation (gather, L2-resident) ----------------
// One wave per node; each lane owns 4 contiguous features (B128 loads),
// neighbor loop unrolled x4 so 4 independent 512B row-gathers are in flight.

__global__ __launch_bounds__(256)
void agg_mean_kernel(const v4f* __restrict__ feat4, const int* __restrict__ rowptr,
                     const int* __restrict__ cols, v4f* __restrict__ agg4) {
    int wave = threadIdx.x >> 5;
    int lane = threadIdx.x & 31;
    int node = blockIdx.x * 8 + wave;   // 8 waves per block
    if (node >= N_NODES) return;

    int s = rowptr[node], e = rowptr[node + 1];
    v4f acc = {};
    int k = s;
    for (; k + 4 <= e; k += 4) {
        int c0 = cols[k], c1 = cols[k + 1], c2 = cols[k + 2], c3 = cols[k + 3];
        v4f r0 = feat4[(size_t)c0 * 32 + lane];
        v4f r1 = feat4[(size_t)c1 * 32 + lane];
        v4f r2 = feat4[(size_t)c2 * 32 + lane];
        v4f r3 = feat4[(size_t)c3 * 32 + lane];
        acc += r0 + r1 + r2 + r3;
    }
    for (; k < e; ++k) {
        acc += feat4[(size_t)cols[k] * 32 + lane];
    }
    float inv = 1.0f / fmaxf((float)(e - s), 1.0f);
    acc *= inv;
    agg4[(size_t)node * 32 + lane] = acc;
}

// ---------------- Fused dual GEMM via V_WMMA_F32_16X16X4_F32 ----------------
// out[m,n] = sum_k agg[m,k]*Wl[n,k] + sum_k x[m,k]*Wr[n,k] + bl[n]
// One wave -> one 16x16 C tile; 8 waves/block cover all 128 output cols.

__global__ __launch_bounds__(256)
void sage_gemm_wmma_kernel(const float* __restrict__ agg, const float* __restrict__ xin,
                           const float* __restrict__ Wl, const float* __restrict__ bl,
                           const float* __restrict__ Wr, float* __restrict__ out) {
    int lane = threadIdx.x & 31;
    int wave = threadIdx.x >> 5;
    int m0 = blockIdx.x * 16;   // N_NODES divisible by 16
    int n0 = wave * 16;
    int l = lane & 15;
    int half = lane >> 4;       // lanes 16-31 carry K+2 (A) / rows K+2 (B)

    const float* ar  = agg + (size_t)(m0 + l) * DIM;
    const float* xr  = xin + (size_t)(m0 + l) * DIM;
    const float* wlr = Wl  + (size_t)(n0 + l) * DIM;
    const float* wrr = Wr  + (size_t)(n0 + l) * DIM;

    v8f c = {};
    #pragma unroll 4
    for (int k0 = 0; k0 < DIM; k0 += 4) {
        int kk = k0 + 2 * half;
        v2f a  = *(const v2f*)(ar + kk);    // A[m0+l][kk..kk+1]
        v2f bw = *(const v2f*)(wlr + kk);   // B[k][n] = Wl[n][k] -> same layout
        c = __builtin_amdgcn_wmma_f32_16x16x4_f32(false, a, false, bw,
                                                  (short)0, c, false, false);
        v2f a2  = *(const v2f*)(xr + kk);
        v2f bw2 = *(const v2f*)(wrr + kk);
        c = __builtin_amdgcn_wmma_f32_16x16x4_f32(false, a2, false, bw2,
                                                  (short)0, c, false, false);
    }

    float bias = bl[n0 + l];
    int baserow = m0 + 8 * half;  // D: vgpr v -> row m0+v (lanes0-15), m0+v+8 (lanes16-31)
    #pragma unroll
    for (int v = 0; v < 8; ++v) {
        out[(size_t)(baserow + v) * DIM + (n0 + l)] = c[v] + bias;
    }
}

// ---------------- BatchNorm (training-mode batch stats) + ReLU ----------------

__global__ __launch_bounds__(128)
void bn_stats_kernel(const float* __restrict__ h, float* __restrict__ stats) {
    int col = threadIdx.x;  // 128 cols, coalesced row reads
    int rowsPer = (N_NODES + gridDim.x - 1) / gridDim.x;
    int r0 = blockIdx.x * rowsPer;
    int r1 = r0 + rowsPer; if (r1 > N_NODES) r1 = N_NODES;
    float s = 0.0f, s2 = 0.0f;
    for (int r = r0; r < r1; ++r) {
        float v = h[(size_t)r * DIM + col];
        s += v; s2 += v * v;
    }
    atomicAdd(&stats[col], s);
    atomicAdd(&stats[DIM + col], s2);
}

__global__ __launch_bounds__(256)
void bn_apply_relu_kernel(const v4f* __restrict__ h4, const float* __restrict__ stats,
                          const float* __restrict__ gamma, const float* __restrict__ beta,
                          v4f* __restrict__ out4) {
    int i = blockIdx.x * blockDim.x + threadIdx.x;   // over N*DIM/4
    if (i >= N_NODES * DIM / 4) return;
    int c0 = (i & 31) * 4;                           // first of 4 consecutive columns
    const float invN = 1.0f / (float)N_NODES;
    v4f v = h4[i];
    v4f y;
    #pragma unroll
    for (int j = 0; j < 4; ++j) {
        int col = c0 + j;
        float m   = stats[col] * invN;
        float var = stats[DIM + col] * invN - m * m;   // biased variance
        float t = (v[j] - m) * rsqrtf(var + BN_EPS) * gamma[col] + beta[col];
        y[j] = fmaxf(t, 0.0f);
    }
    out4[i] = y;
}

// ---------------- Launch ----------------

extern "C" void kernel_launch(void* const* d_in, const int* in_sizes, int n_in,
                              void* d_out, int out_size, void* d_ws, size_t ws_size,
                              hipStream_t stream) {
    const float*     x   = (const float*)d_in[0];
    const long long* ei  = (const long long*)d_in[1];   // int64 [2,E]
    const float* W1l = (const float*)d_in[2];
    const float* b1l = (const float*)d_in[3];
    const float* W1r = (const float*)d_in[4];
    const float* g1  = (const float*)d_in[5];
    const float* be1 = (const float*)d_in[6];
    const float* W2l = (const float*)d_in[7];
    const float* b2l = (const float*)d_in[8];
    const float* W2r = (const float*)d_in[9];
    const float* g2  = (const float*)d_in[10];
    const float* be2 = (const float*)d_in[11];
    float* out = (float*)d_out;

    char* ws = (char*)d_ws;
    size_t off = 0;
    auto salloc = [&](size_t bytes) -> void* {
        void* p = ws + off;
        off += (bytes + 255) & ~(size_t)255;
        return p;
    };
    int*   deg    = (int*)  salloc((size_t)N_NODES * 4);
    int*   rowptr = (int*)  salloc((size_t)(N_NODES + 1) * 4);
    int*   cursor = (int*)  salloc((size_t)N_NODES * 4);
    int*   cols   = (int*)  salloc((size_t)N_EDGES * 4);
    float* aggb   = (float*)salloc((size_t)N_NODES * DIM * 4);
    float* lin    = (float*)salloc((size_t)N_NODES * DIM * 4);
    float* stats  = (float*)salloc(256 * 4);

    const long long* srcp = ei;            // edge_index[0]
    const long long* dstp = ei + N_EDGES;  // edge_index[1]

    const int TB = 256;
    // Build CSR once; reused by both layers.
    zero_i32_kernel<<<(N_NODES + TB - 1) / TB, TB, 0, stream>>>(deg, N_NODES);
    count_deg_kernel<<<(N_EDGES + TB - 1) / TB, TB, 0, stream>>>(dstp, deg);
    exscan_kernel<<<1, 1024, 0, stream>>>(deg, rowptr, N_NODES);
    init_cursor_kernel<<<(N_NODES + TB - 1) / TB, TB, 0, stream>>>(rowptr, cursor);
    scatter_edges_kernel<<<(N_EDGES + TB - 1) / TB, TB, 0, stream>>>(srcp, dstp, cursor, cols);

    const int AGG_BLOCKS = (N_NODES + 7) / 8;          // 8 nodes (waves) per block
    const int BN_ELEMS4  = N_NODES * DIM / 4;

    // ---- Layer 1: h1 = relu(bn(sage(x)))  (h1 lives in d_out temporarily) ----
    agg_mean_kernel<<<AGG_BLOCKS, 256, 0, stream>>>((const v4f*)x, rowptr, cols, (v4f*)aggb);
    sage_gemm_wmma_kernel<<<N_NODES / 16, 256, 0, stream>>>(aggb, x, W1l, b1l, W1r, lin);
    zero_f32_kernel<<<1, 256, 0, stream>>>(stats, 256);
    bn_stats_kernel<<<400, 128, 0, stream>>>(lin, stats);
    bn_apply_relu_kernel<<<(BN_ELEMS4 + TB - 1) / TB, TB, 0, stream>>>(
        (const v4f*)lin, stats, g1, be1, (v4f*)out);

    // ---- Layer 2: out = relu(bn(sage(h1))) ----
    agg_mean_kernel<<<AGG_BLOCKS, 256, 0, stream>>>((const v4f*)out, rowptr, cols, (v4f*)aggb);
    sage_gemm_wmma_kernel<<<N_NODES / 16, 256, 0, stream>>>(aggb, out, W2l, b2l, W2r, lin);
    zero_f32_kernel<<<1, 256, 0, stream>>>(stats, 256);
    bn_stats_kernel<<<400, 128, 0, stream>>>(lin, stats);
    bn_apply_relu_kernel<<<(BN_ELEMS4 + TB - 1) / TB, TB, 0, stream>>>(
        (const v4f*)lin, stats, g2, be2, (v4f*)out);
}